// YOLO_postprocess_76656576299741
// MI455X (gfx1250) — compile-verified
//
#include <hip/hip_runtime.h>
#include <hip/hip_bf16.h>
#include <stddef.h>
#include <stdint.h>

// ---------------- constants ----------------
#define N_ANCH   8400
#define NWORDS   263            // ceil(8400/32) valid mask words per row
#define MSTRIDE  264            // padded row stride (1056 B, 16B aligned for async B128)
#define SORT_M   16384          // next pow2 >= 8400
#define ROWT     525            // 8400/16 row tiles
#define SCORE_THR 0.5f
#define IOU_THR   0.5f

// async-prefetch scan parameters
#define NSLOT    32             // LDS ring slots (> PFD)
#define SLOT_W   384            // words per slot (1536 B = 3 x 512 B wave fetches)
#define PFD      16             // prefetch depth in rows (3*PFD = 48 <= 63 ASYNCcnt max)

// ---------------- ws layout (in 4-byte words) ----------------
// 0      ux1 | 8400 uy1 | 16800 ux2 | 25200 uy2 | 33600 us | 42000 ukey
// 50400  sx1 | 58800 sy1 | 67200 sx2 | 75600 sy2 | 84000 ss | 92400 area
// 100800 validWords (u32 x263) | 101063 keepWords (u32 x263)
// 101376 mask (u32 x 8400*264 + 384 pad)  => total ~9.3 MB
#define WS_UX1   0
#define WS_UY1   8400
#define WS_UX2   16800
#define WS_UY2   25200
#define WS_US    33600
#define WS_UKEY  42000
#define WS_SX1   50400
#define WS_SY1   58800
#define WS_SX2   67200
#define WS_SY2   75600
#define WS_SS    84000
#define WS_AREA  92400
#define WS_VW    100800
#define WS_KW    101063
#define WS_MASK  101376

typedef __attribute__((ext_vector_type(2))) float v2f;
typedef __attribute__((ext_vector_type(8))) float v8f;

// ---------------- kernel A: decode boxes + sort keys ----------------
__global__ void yolo_prep_kernel(const float* __restrict__ raw, float* __restrict__ ws) {
    int i = blockIdx.x * blockDim.x + threadIdx.x;
    if (i >= N_ANCH) return;
    float cx = raw[i];
    float cy = raw[N_ANCH + i];
    float w  = raw[2 * N_ANCH + i];
    float h  = raw[3 * N_ANCH + i];
    float s  = raw[4 * N_ANCH + i];
    ws[WS_UX1 + i] = cx - 0.5f * w;
    ws[WS_UY1 + i] = cy - 0.5f * h;
    ws[WS_UX2 + i] = cx + 0.5f * w;
    ws[WS_UY2 + i] = cy + 0.5f * h;
    ws[WS_US  + i] = s;
    // scores are uniform[0,1]; -1.0f sorts below every valid score (acts as -inf)
    ws[WS_UKEY + i] = (s >= SCORE_THR) ? s : -1.0f;
}

// ---------------- kernel B: single-WG LDS bitonic argsort + gather ----------------
// 128 KB static LDS: legal on CDNA5 (320 KB / workgroup).
__global__ __launch_bounds__(1024, 1) void yolo_sort_kernel(float* __restrict__ ws) {
    __shared__ float skey[SORT_M];
    __shared__ int   sidx[SORT_M];
    const int tid = threadIdx.x;

    for (int m = tid; m < SORT_M; m += 1024) {
        skey[m] = (m < N_ANCH) ? ws[WS_UKEY + m] : -2.0f;
        sidx[m] = m;
    }
    __syncthreads();

    // descending bitonic sort of (key, idx)
    for (int k = 2; k <= SORT_M; k <<= 1) {
        for (int j = k >> 1; j > 0; j >>= 1) {
            for (int m = tid; m < SORT_M; m += 1024) {
                int ixj = m ^ j;
                if (ixj > m) {
                    bool desc = ((m & k) == 0);
                    float a = skey[m], b = skey[ixj];
                    if ((a < b) == desc) {
                        skey[m] = b; skey[ixj] = a;
                        int t = sidx[m]; sidx[m] = sidx[ixj]; sidx[ixj] = t;
                    }
                }
            }
            __syncthreads();
        }
    }

    // gather sorted boxes / scores, precompute areas
    for (int m = tid; m < N_ANCH; m += 1024) {
        int o = sidx[m];
        float x1 = ws[WS_UX1 + o], y1 = ws[WS_UY1 + o];
        float x2 = ws[WS_UX2 + o], y2 = ws[WS_UY2 + o];
        ws[WS_SX1 + m] = x1;
        ws[WS_SY1 + m] = y1;
        ws[WS_SX2 + m] = x2;
        ws[WS_SY2 + m] = y2;
        ws[WS_SS  + m] = ws[WS_US + o];
        ws[WS_AREA + m] = (x2 - x1) * (y2 - y1);
    }

    // valid bitmask from sorted keys (key >= 0.5 <=> valid), wave32 ballot
    unsigned* vw = (unsigned*)(ws + WS_VW);
    for (int m = tid; m < NWORDS * 32; m += 1024) {
        bool pred = (m < N_ANCH) && (skey[m] >= SCORE_THR);
        unsigned b = (unsigned)__ballot(pred);
        if ((m & 31) == 0) vw[m >> 5] = b;
    }
}

// ---------------- kernel C: suppression bitmask, 16x32 tile per wave ----------------
// union term area_i + area_j computed as a rank-2 outer sum via V_WMMA_F32_16X16X4_F32;
// D-matrix layout (lane = col n, VGPR r = row r+8*hi) matches the per-lane IoU work split.
__global__ void yolo_mask_kernel(const float* __restrict__ ws, unsigned* __restrict__ mask) {
    const int lane = threadIdx.x;                           // 0..31
    const int rt   = blockIdx.y * blockDim.y + threadIdx.y; // row tile (16 rows)
    const int w    = blockIdx.x;                            // 32-col word index
    if (rt >= ROWT) return;                                 // wave-uniform exit

    const float* sx1  = ws + WS_SX1;
    const float* sy1  = ws + WS_SY1;
    const float* sx2  = ws + WS_SX2;
    const float* sy2  = ws + WS_SY2;
    const float* area = ws + WS_AREA;

    // rows: lanes 0..15 own rows rt*16 + lane (lanes 16..31 mirror for shuffles)
    const int rl = rt * 16 + (lane & 15);
    float rx1 = sx1[rl], ry1 = sy1[rl], rx2 = sx2[rl], ry2 = sy2[rl];
    float rar = area[rl];

    // cols: lane owns absolute col w*32 + lane (clamped; OOB masked in predicate)
    int c = w * 32 + lane;
    int cc = (c < N_ANCH) ? c : (N_ANCH - 1);
    float cx1 = sx1[cc], cy1 = sy1[cc], cx2 = sx2[cc], cy2 = sy2[cc];
    float car = area[cc];

    const bool lo = (lane < 16);
    const int  n  = lane & 15;        // tile-local column this lane computes
    const int  hi = lane >> 4;        // which 8-row half this lane computes

    // A 16x4 f32: lanes 0-15 = (K0,K1) = (area_row, 1); lanes 16-31 = (K2,K3) = 0
    v2f a; a.x = lo ? rar : 0.0f; a.y = lo ? 1.0f : 0.0f;
    // B 4x16 f32, tile0: lanes 0-15 = (K0,K1) = (1, area_col); lanes 16-31 = 0
    v2f b0; b0.x = lo ? 1.0f : 0.0f; b0.y = lo ? car : 0.0f;
    // tile1 B: columns 16..31 of the word
    float car1 = __shfl(car, n + 16);
    v2f b1; b1.x = lo ? 1.0f : 0.0f; b1.y = lo ? car1 : 0.0f;

    v8f cz = {0.f, 0.f, 0.f, 0.f, 0.f, 0.f, 0.f, 0.f};
    v8f d0 = __builtin_amdgcn_wmma_f32_16x16x4_f32(false, a, false, b0, (short)0, cz, false, false);
    v8f d1 = __builtin_amdgcn_wmma_f32_16x16x4_f32(false, a, false, b1, (short)0, cz, false, false);

    // col box data for this lane's columns in each tile
    float t0x1 = __shfl(cx1, n),      t0y1 = __shfl(cy1, n);
    float t0x2 = __shfl(cx2, n),      t0y2 = __shfl(cy2, n);
    float t1x1 = __shfl(cx1, n + 16), t1y1 = __shfl(cy1, n + 16);
    float t1x2 = __shfl(cx2, n + 16), t1y2 = __shfl(cy2, n + 16);
    const int c0 = w * 32 + n;
    const int c1 = c0 + 16;

#pragma unroll
    for (int r = 0; r < 8; r++) {
        const int src = r + 8 * hi;                   // lane holding this row's box
        float ax1 = __shfl(rx1, src), ay1 = __shfl(ry1, src);
        float ax2 = __shfl(rx2, src), ay2 = __shfl(ry2, src);
        const int rowAbs = rt * 16 + r + 8 * hi;

        // tile0: intersection; union from WMMA outer-sum
        float dx0 = fmaxf(fminf(ax2, t0x2) - fmaxf(ax1, t0x1), 0.0f);
        float dy0 = fmaxf(fminf(ay2, t0y2) - fmaxf(ay1, t0y1), 0.0f);
        float in0 = dx0 * dy0;
        float un0 = d0[r] - in0;
        bool  p0  = (in0 > IOU_THR * fmaxf(un0, 1e-9f)) && (c0 > rowAbs) && (c0 < N_ANCH);

        // tile1
        float dx1 = fmaxf(fminf(ax2, t1x2) - fmaxf(ax1, t1x1), 0.0f);
        float dy1 = fmaxf(fminf(ay2, t1y2) - fmaxf(ay1, t1y1), 0.0f);
        float in1 = dx1 * dy1;
        float un1 = d1[r] - in1;
        bool  p1  = (in1 > IOU_THR * fmaxf(un1, 1e-9f)) && (c1 > rowAbs) && (c1 < N_ANCH);

        // ballot: bits 0-15 = row (rt*16+r) cols 0-15 ; bits 16-31 = row (+8) cols 0-15
        unsigned bt0 = (unsigned)__ballot(p0);
        unsigned bt1 = (unsigned)__ballot(p1);
        unsigned wa = (bt0 & 0xFFFFu) | ((bt1 & 0xFFFFu) << 16); // row rt*16+r
        unsigned wb = (bt0 >> 16)     | (bt1 & 0xFFFF0000u);     // row rt*16+r+8
        if (lane == 0) mask[(size_t)(rt * 16 + r) * MSTRIDE + w] = wa;
        if (lane == 1) mask[(size_t)(rt * 16 + r + 8) * MSTRIDE + w] = wb;
    }
}

// ---------------- kernel D: serial NMS scan, async-prefetched mask rows ----------------
// Single wave32. Mask rows are streamed unconditionally into an LDS ring with
// global_load_async_to_lds_b128 (ASYNCcnt, in-order completion), PFD rows ahead,
// so the serial keep/suppress chain never waits on L2 latency.
// Keep bits live in registers: lane owns words {lane, lane+32, ..., lane+256}.
__global__ void yolo_scan_kernel(float* __restrict__ ws, const unsigned* __restrict__ mask) {
    __shared__ unsigned abuf[NSLOT * SLOT_W];   // 48 KB ring
    const int lane = threadIdx.x;               // 32 threads
    const unsigned* vw = (const unsigned*)(ws + WS_VW);
    unsigned* kw = (unsigned*)(ws + WS_KW);

    const unsigned ldsbase = (unsigned)(uintptr_t)(void*)&abuf[0]; // LDS aperture: low 32 bits = LDS addr
    const unsigned vbyte   = (unsigned)lane * 16u;

    unsigned keep[9];
#pragma unroll
    for (int r = 0; r < 9; r++) {
        int w2 = r * 32 + lane;
        keep[r] = (w2 < NWORDS) ? vw[w2] : 0u;
    }

    // prologue: prefetch rows 0..PFD-1 (3 async B128 per row: 3*512B >= 1056B row)
    for (int p = 0; p < PFD; ++p) {
        unsigned long long gbase = (unsigned long long)(uintptr_t)(mask + (size_t)p * MSTRIDE);
        unsigned ldst = ldsbase + (unsigned)(p & (NSLOT - 1)) * (SLOT_W * 4u) + vbyte;
        asm volatile(
            "global_load_async_to_lds_b128 %0, %1, %2 offset:0\n\t"
            "global_load_async_to_lds_b128 %0, %1, %2 offset:512\n\t"
            "global_load_async_to_lds_b128 %0, %1, %2 offset:1024"
            :: "v"(ldst), "v"(vbyte), "s"(gbase) : "memory");
    }

    for (int i = 0; i < N_ANCH; i++) {
        // wait until row i's 3 loads have landed in LDS (in-order completion):
        // steady state has 3*PFD=48 outstanding; asynccnt<=45 => oldest row done.
        if (i + PFD < N_ANCH) {
            asm volatile("s_wait_asynccnt 45" ::: "memory");
        } else {
            asm volatile("s_wait_asynccnt 0" ::: "memory");
        }
        // refill the ring PFD rows ahead
        if (i + PFD < N_ANCH) {
            int p = i + PFD;
            unsigned long long gbase = (unsigned long long)(uintptr_t)(mask + (size_t)p * MSTRIDE);
            unsigned ldst = ldsbase + (unsigned)(p & (NSLOT - 1)) * (SLOT_W * 4u) + vbyte;
            asm volatile(
                "global_load_async_to_lds_b128 %0, %1, %2 offset:0\n\t"
                "global_load_async_to_lds_b128 %0, %1, %2 offset:512\n\t"
                "global_load_async_to_lds_b128 %0, %1, %2 offset:1024"
                :: "v"(ldst), "v"(vbyte), "s"(gbase) : "memory");
        }

        const int w = i >> 5, bit = i & 31;
        const int reg = w >> 5, owner = w & 31;
        unsigned sel;
        switch (reg) {
            case 0: sel = keep[0]; break;
            case 1: sel = keep[1]; break;
            case 2: sel = keep[2]; break;
            case 3: sel = keep[3]; break;
            case 4: sel = keep[4]; break;
            case 5: sel = keep[5]; break;
            case 6: sel = keep[6]; break;
            case 7: sel = keep[7]; break;
            default: sel = keep[8]; break;
        }
        unsigned word = (unsigned)__shfl((int)sel, owner); // wave-uniform broadcast
        if ((word >> bit) & 1u) {
            const unsigned* row = &abuf[(i & (NSLOT - 1)) * SLOT_W];
#pragma unroll
            for (int r = 0; r < 9; r++) {
                int w2 = r * 32 + lane;
                if (w2 < NWORDS) keep[r] &= ~row[w2]; // mask already restricted to j>i
            }
        }
    }

#pragma unroll
    for (int r = 0; r < 9; r++) {
        int w2 = r * 32 + lane;
        if (w2 < NWORDS) kw[w2] = keep[r];
    }
}

// ---------------- kernel E: write [x1,y1,x2,y2,score] * keep ----------------
__global__ void yolo_out_kernel(const float* __restrict__ ws, float* __restrict__ out) {
    int i = blockIdx.x * blockDim.x + threadIdx.x;
    if (i >= N_ANCH) return;
    const unsigned* kw = (const unsigned*)(ws + WS_KW);
    float k = ((kw[i >> 5] >> (i & 31)) & 1u) ? 1.0f : 0.0f;
    out[i * 5 + 0] = ws[WS_SX1 + i] * k;
    out[i * 5 + 1] = ws[WS_SY1 + i] * k;
    out[i * 5 + 2] = ws[WS_SX2 + i] * k;
    out[i * 5 + 3] = ws[WS_SY2 + i] * k;
    out[i * 5 + 4] = ws[WS_SS  + i] * k;
}

extern "C" void kernel_launch(void* const* d_in, const int* in_sizes, int n_in,
                              void* d_out, int out_size, void* d_ws, size_t ws_size,
                              hipStream_t stream) {
    (void)in_sizes; (void)n_in; (void)out_size; (void)ws_size;
    const float* raw = (const float*)d_in[0];
    float* ws = (float*)d_ws;
    unsigned* mask = (unsigned*)(ws + WS_MASK);   // 8400*264 words + over-fetch pad
    float* out = (float*)d_out;

    yolo_prep_kernel<<<(N_ANCH + 255) / 256, 256, 0, stream>>>(raw, ws);
    yolo_sort_kernel<<<1, 1024, 0, stream>>>(ws);
    {
        dim3 blk(32, 8);                       // 8 wave32 per block, one row-tile each
        dim3 grd(NWORDS, (ROWT + 7) / 8);
        yolo_mask_kernel<<<grd, blk, 0, stream>>>(ws, mask);
    }
    yolo_scan_kernel<<<1, 32, 0, stream>>>(ws, mask);
    yolo_out_kernel<<<(N_ANCH + 255) / 256, 256, 0, stream>>>(ws, out);
}